// MeanPooling_28544352649940
// MI455X (gfx1250) — compile-verified
//
#include <hip/hip_runtime.h>

typedef __attribute__((ext_vector_type(4))) float v4f;

#ifndef SEG_D
#define SEG_D 64
#endif

// ---------------------------------------------------------------------------
// Kernel 1: zero the sum accumulator (d_out) and the count array (d_ws).
// Harness poisons d_out/d_ws once and never re-poisons between replays, so
// every kernel_launch re-zeros (deterministic).
// ---------------------------------------------------------------------------
__global__ __launch_bounds__(256) void seg_zero_kernel(float* __restrict__ out,
                                                       int out_n4,  // out_size/4
                                                       float* __restrict__ cnt,
                                                       int cnt_n) {
  int i = blockIdx.x * blockDim.x + threadIdx.x;
  if (i < out_n4) {
    v4f z = {0.f, 0.f, 0.f, 0.f};
    ((v4f*)out)[i] = z;
  }
  if (i < cnt_n) cnt[i] = 0.f;
}

// ---------------------------------------------------------------------------
// Kernel 2: scatter-add. 16 threads per row; thread `sub` owns columns
// {sub, sub+16, sub+32, sub+48}.
//
// Column-interleaved ownership makes each global_atomic_add_f32 instruction's
// 16 lanes (one row) hit 16 CONSECUTIVE dwords = exactly one 64B cacheline,
// so every accumulator line is RMW'd once per row at L2 (vs 4x with a
// thread-contiguous float4 layout). Atomic line-transactions drop 4x.
//
// Loads: 4 NT b32 loads; per instruction a wave reads 2 contiguous 64B
// blocks (rows r, r+1 at the same column offset) - fully coalesced, same
// total traffic as one b128. x is streamed once -> NT keeps it from evicting
// the 25.6MB accumulator that must stay L2-resident.
// ---------------------------------------------------------------------------
__global__ __launch_bounds__(256) void seg_scatter_kernel(
    const float* __restrict__ x, const int* __restrict__ index,
    float* __restrict__ sums, float* __restrict__ cnt, int nrows) {
  unsigned int gid = blockIdx.x * blockDim.x + threadIdx.x;
  int row = (int)(gid >> 4);
  if (row >= nrows) return;
  int sub = (int)(gid & 15u);

  int seg = index[row];

  const float* xr = x + (unsigned int)row * SEG_D;
  float a0 = __builtin_nontemporal_load(xr + sub);
  float a1 = __builtin_nontemporal_load(xr + sub + 16);
  float a2 = __builtin_nontemporal_load(xr + sub + 32);
  float a3 = __builtin_nontemporal_load(xr + sub + 48);

  float* dst = sums + (unsigned int)seg * SEG_D;
  atomicAdd(dst + sub,      a0);  // lanes 0-15: one 64B line of seg(row0)
  atomicAdd(dst + sub + 16, a1);
  atomicAdd(dst + sub + 32, a2);
  atomicAdd(dst + sub + 48, a3);

  if (sub == 0) atomicAdd(cnt + seg, 1.0f);
}

// ---------------------------------------------------------------------------
// Kernel 3: divide by clamped count (scatter_mean: empty segment -> 0).
// One thread per float4; 16 float4 per segment row.
// ---------------------------------------------------------------------------
__global__ __launch_bounds__(256) void seg_finalize_kernel(
    float* __restrict__ out, const float* __restrict__ cnt, int total4) {
  int i = blockIdx.x * blockDim.x + threadIdx.x;
  if (i >= total4) return;
  int seg = i >> 4;  // 64 floats = 16 float4 per segment
  float c = cnt[seg];
  float inv = 1.0f / fmaxf(c, 1.0f);
  v4f v = ((v4f*)out)[i];
  v.x *= inv; v.y *= inv; v.z *= inv; v.w *= inv;
  ((v4f*)out)[i] = v;
}

extern "C" void kernel_launch(void* const* d_in, const int* in_sizes, int n_in,
                              void* d_out, int out_size, void* d_ws, size_t ws_size,
                              hipStream_t stream) {
  const float* x   = (const float*)d_in[0];
  const int*   idx = (const int*)d_in[1];   // harness passes integer inputs as int32
  float* out = (float*)d_out;
  float* cnt = (float*)d_ws;                // S floats (400 KB for S=100k)

  const int nrows = in_sizes[1];            // N (one index entry per row)
  const int S = out_size / SEG_D;           // number of segments
  const int out_n4 = out_size >> 2;         // float4 count of output

  // 1) zero sums + counts
  {
    int zthreads = out_n4 > S ? out_n4 : S;
    int blocks = (zthreads + 255) / 256;
    seg_zero_kernel<<<blocks, 256, 0, stream>>>(out, out_n4, cnt, S);
  }

  // 2) scatter-add (16 threads per row)
  {
    long long total = (long long)nrows * 16;
    int blocks = (int)((total + 255) / 256);
    seg_scatter_kernel<<<blocks, 256, 0, stream>>>(x, idx, out, cnt, nrows);
  }

  // 3) finalize: divide by clamped counts
  {
    int total4 = S * 16;
    int blocks = (total4 + 255) / 256;
    seg_finalize_kernel<<<blocks, 256, 0, stream>>>(out, cnt, total4);
  }
}